// ContrastiveLoss_33792802685631
// MI455X (gfx1250) — compile-verified
//
#include <hip/hip_runtime.h>
#include <hip/hip_bf16.h>

typedef float v2f __attribute__((ext_vector_type(2)));
typedef float v8f __attribute__((ext_vector_type(8)));

#define NROWS 8192      // 2B rows per term
#define DIM   256       // embedding dim
#define HALF  4096      // B
#define STRIP 128       // rows per workgroup (8 waves x 16 rows)
#define LDS_STRIDE 260  // 256 + 4 pad floats -> conflict-free b64 B reads
#define TILE_FLOATS (16 * LDS_STRIDE)
#define NTILES (NROWS / 16)

// Low 32 bits of a generic pointer to __shared__ == LDS byte offset
// (generic LDS addr = {SHARED_BASE, offset32}; HW discards upper bits).
__device__ __forceinline__ unsigned lds_off(const void* p) {
    return (unsigned)(unsigned long long)p;
}

// Async global->LDS copy of 64 contiguous bytes (4 x b128) for this thread.
// IOFFSET is added to BOTH the LDS and global addresses (ISA 15.18.3), so a
// single base pair + immediate offsets covers the whole 64B. Tracked by
// ASYNCcnt; no VGPR destinations, overlaps with WMMA compute.
__device__ __forceinline__ void async_copy64(unsigned lds_dst, const float* gsrc) {
    unsigned long long ga = (unsigned long long)gsrc;
    asm volatile("global_load_async_to_lds_b128 %0, %1, off"            :: "v"(lds_dst), "v"(ga) : "memory");
    asm volatile("global_load_async_to_lds_b128 %0, %1, off offset:16"  :: "v"(lds_dst), "v"(ga) : "memory");
    asm volatile("global_load_async_to_lds_b128 %0, %1, off offset:32"  :: "v"(lds_dst), "v"(ga) : "memory");
    asm volatile("global_load_async_to_lds_b128 %0, %1, off offset:48"  :: "v"(lds_dst), "v"(ga) : "memory");
}

__device__ __forceinline__ void wait_async0() {
    asm volatile("s_wait_asynccnt 0x0" ::: "memory");
}

// ---------------------------------------------------------------------------
// Phase 1: row-normalize teacher/student into the two per-term feature
// matrices F0 = [t[0]; s[1]], F1 = [t[1]; s[0]] (each 8192 x 256, f32).
// ---------------------------------------------------------------------------
__global__ __launch_bounds__(256)
void cl_normalize_kernel(const float* __restrict__ teacher,
                         const float* __restrict__ student,
                         float* __restrict__ F0, float* __restrict__ F1)
{
    int wave = blockIdx.x * 8 + (threadIdx.x >> 5);   // 0..16383
    int lane = threadIdx.x & 31;

    const float* src;
    float* dst;
    if (wave < NROWS) {                 // teacher row
        int row = wave;
        src = teacher + (size_t)row * DIM;
        dst = (row < HALF) ? (F0 + (size_t)row * DIM)
                           : (F1 + (size_t)(row - HALF) * DIM);
    } else {                            // student row
        int row = wave - NROWS;
        src = student + (size_t)row * DIM;
        dst = (row >= HALF) ? (F0 + (size_t)row * DIM)
                            : (F1 + (size_t)(row + HALF) * DIM);
    }

    float v[8];
    float ss = 0.f;
#pragma unroll
    for (int i = 0; i < 8; ++i) {
        v[i] = src[lane + 32 * i];
        ss += v[i] * v[i];
    }
#pragma unroll
    for (int m = 1; m < 32; m <<= 1) ss += __shfl_xor(ss, m, 32);

    float nrm = fmaxf(sqrtf(ss), 1e-8f);
    float inv = 1.0f / nrm;
#pragma unroll
    for (int i = 0; i < 8; ++i) dst[lane + 32 * i] = v[i] * inv;
}

// ---------------------------------------------------------------------------
// Phase 2: fused sim = F F^T (f32 WMMA 16x16x4, K=256) + shifted logsumexp.
// Double-buffered LDS column tiles fed by async global->LDS copies; B-feed
// pipelined 4 ds_load_2addr_b64 ahead of the WMMA chain via
// sched_group_barrier (DS-read group -> WMMA group interleave).
// ---------------------------------------------------------------------------
__global__ __launch_bounds__(256)
void cl_simloss_kernel(const float* __restrict__ F0, const float* __restrict__ F1,
                       const float* __restrict__ wp, const float* __restrict__ bp_,
                       float* __restrict__ rowloss)
{
    __shared__ __align__(16) float tile[2][TILE_FLOATS];

    const int term  = blockIdx.x >> 6;   // 0..1
    const int strip = blockIdx.x & 63;   // 0..63
    const float* __restrict__ F = term ? F1 : F0;

    const int tid  = threadIdx.x;
    const int wave = tid >> 5;
    const int lane = tid & 31;
    const int l16  = lane & 15;
    const int hi   = lane >> 4;          // lane half: selects K pair (+2)

    const float w0    = wp[0];
    const float b0    = bp_[0];
    const float shift = fabsf(w0) + b0;  // logits bounded by |w|*1 + b

    const int rowbase = strip * STRIP + wave * 16;

    // --- A operands: ISA 16x4 f32 layout, lane<16 -> K={4k,4k+1}, else +2 ---
    v2f a[64];
    {
        const float* ap = F + (size_t)(rowbase + l16) * DIM + hi * 2;
#pragma unroll
        for (int kk = 0; kk < 64; ++kk)
            a[kk] = *(const v2f*)(ap + 4 * kk);
    }

    // staging geometry: thread -> (tile row, 16-float chunk), 64B per thread
    const int srow   = tid >> 4;
    const int schunk = tid & 15;
    const unsigned myLds0 = lds_off(&tile[0][srow * LDS_STRIDE + schunk * 16]);
    const unsigned myLds1 = lds_off(&tile[1][srow * LDS_STRIDE + schunk * 16]);
    const float* gstage = F + (size_t)srow * DIM + schunk * 16;  // + j*16*DIM

    float ssum[8], lpos[8];
#pragma unroll
    for (int i = 0; i < 8; ++i) { ssum[i] = 0.f; lpos[i] = 0.f; }

    // prologue: stage tile 0 into buffer 0
    async_copy64(myLds0, gstage);
    wait_async0();
    __syncthreads();

    for (int j = 0; j < NTILES; ++j) {
        const int cur = j & 1;
        // branchless prefetch of tile j+1 into the other buffer (last iter
        // redundantly re-stages tile NTILES-1 into the dead buffer; keeps the
        // loop body a single scheduling region for the group barriers below)
        {
            int jn = j + 1;
            jn = (jn < NTILES) ? jn : (NTILES - 1);
            async_copy64(cur ? myLds0 : myLds1,
                         gstage + (size_t)jn * 16 * DIM);
        }

        v8f cacc;
#pragma unroll
        for (int i = 0; i < 8; ++i) cacc[i] = 0.f;

        const float* bbase = &tile[cur][0] + l16 * LDS_STRIDE + hi * 2;

        // K=256 as 8 chunks of 8 k-steps; register double-buffer the B feed
        v2f bb[2][8];
#pragma unroll
        for (int t = 0; t < 8; ++t)
            bb[0][t] = *(const v2f*)(bbase + 4 * t);

#pragma unroll
        for (int ch = 0; ch < 8; ++ch) {
            const int cb = ch & 1;
            if (ch < 7) {
#pragma unroll
                for (int t = 0; t < 8; ++t)
                    bb[cb ^ 1][t] = *(const v2f*)(bbase + 4 * ((ch + 1) * 8 + t));
            }
#pragma unroll
            for (int t = 0; t < 8; ++t)
                cacc = __builtin_amdgcn_wmma_f32_16x16x4_f32(
                           false, a[ch * 8 + t], false, bb[cb][t],
                           (short)0, cacc, false, false);
        }

        // Static issue pattern for this region: keep one 4-load DS-read group
        // (8 b64 values = next chunk) in flight ahead of each 8-WMMA group.
        // 0x100 = DS read, 0x008 = MFMA/WMMA.
        __builtin_amdgcn_sched_group_barrier(0x100, 4, 0);   // chunk 0 loads
#pragma unroll
        for (int c = 0; c < 7; ++c) {
            __builtin_amdgcn_sched_group_barrier(0x100, 4, 0); // chunk c+1 loads
            __builtin_amdgcn_sched_group_barrier(0x008, 8, 0); // chunk c WMMAs
        }
        __builtin_amdgcn_sched_group_barrier(0x008, 8, 0);     // chunk 7 WMMAs

        // fused epilogue: shifted exp-sum, diagonal mask, positive pick
        const int gcol = j * 16 + l16;
#pragma unroll
        for (int v = 0; v < 8; ++v) {
            const int grow = rowbase + v + hi * 8;
            const float l  = w0 * cacc[v] + b0;
            const float e  = (gcol == grow) ? 0.f : __expf(l - shift);
            ssum[v] += e;
            lpos[v] += (gcol == (grow ^ HALF)) ? l : 0.f;
        }

        wait_async0();      // tile j+1 resident
        __syncthreads();    // everyone done reading tile j / writing j+1
    }

    // reduce across the 16 lanes that share each row
#pragma unroll
    for (int v = 0; v < 8; ++v) {
        float s = ssum[v], p = lpos[v];
#pragma unroll
        for (int m = 1; m < 16; m <<= 1) {
            s += __shfl_xor(s, m, 32);
            p += __shfl_xor(p, m, 32);
        }
        ssum[v] = s; lpos[v] = p;
    }

    if (l16 == 0) {
#pragma unroll
        for (int v = 0; v < 8; ++v) {
            const int grow = rowbase + v + hi * 8;
            rowloss[term * NROWS + grow] = lpos[v] - shift - __logf(ssum[v]);
        }
    }
}

// ---------------------------------------------------------------------------
// Phase 3: deterministic single-block reduction to the scalar loss.
// out = -(1/0.07) * sum(rowloss) / 16384
// ---------------------------------------------------------------------------
__global__ __launch_bounds__(256)
void cl_reduce_kernel(const float* __restrict__ rowloss, float* __restrict__ out)
{
    __shared__ float red[256];
    float s = 0.f;
    for (int i = threadIdx.x; i < 2 * NROWS; i += 256) s += rowloss[i];
    red[threadIdx.x] = s;
    __syncthreads();
    for (int off = 128; off > 0; off >>= 1) {
        if (threadIdx.x < off) red[threadIdx.x] += red[threadIdx.x + off];
        __syncthreads();
    }
    if (threadIdx.x == 0)
        out[0] = -(1.0f / 0.07f) * red[0] / (float)(2 * NROWS);
}

extern "C" void kernel_launch(void* const* d_in, const int* in_sizes, int n_in,
                              void* d_out, int out_size, void* d_ws, size_t ws_size,
                              hipStream_t stream) {
    (void)in_sizes; (void)n_in; (void)out_size; (void)ws_size;
    const float* student = (const float*)d_in[0];
    const float* teacher = (const float*)d_in[1];
    const float* w       = (const float*)d_in[2];
    const float* b       = (const float*)d_in[3];
    float* out           = (float*)d_out;

    float* F0      = (float*)d_ws;                 // 8192*256 f32 (8 MB)
    float* F1      = F0 + (size_t)NROWS * DIM;     // 8192*256 f32 (8 MB)
    float* rowloss = F1 + (size_t)NROWS * DIM;     // 16384 f32

    cl_normalize_kernel<<<dim3((2 * NROWS) / 8), dim3(256), 0, stream>>>(
        teacher, student, F0, F1);
    cl_simloss_kernel<<<dim3(2 * (NROWS / STRIP)), dim3(256), 0, stream>>>(
        F0, F1, w, b, rowloss);
    cl_reduce_kernel<<<dim3(1), dim3(256), 0, stream>>>(rowloss, out);
}